// DeformConvNet_55387898249789
// MI455X (gfx1250) — compile-verified
//
#include <hip/hip_runtime.h>
#include <hip/hip_bf16.h>

typedef __attribute__((ext_vector_type(2))) float v2f;
typedef __attribute__((ext_vector_type(8))) float v8f;
typedef int gv4i __attribute__((vector_size(16)));   // matches builtin param type

#define NB    16      // batch
#define CIN   256     // input channels
#define COUT  256     // deform output channels
#define KTAPS 9
#define KTOT  (CIN * KTAPS)   // 2304 GEMM K dim
#define KS    64              // K slab staged in LDS per iteration

#define AS1 __attribute__((address_space(1)))
#define AS3 __attribute__((address_space(3)))

#if defined(__AMDGCN__) && \
    __has_builtin(__builtin_amdgcn_global_load_async_to_lds_b128) && \
    __has_builtin(__builtin_amdgcn_s_wait_asynccnt)
#define HAVE_ASYNC_LDS 1
#else
#define HAVE_ASYNC_LDS 0
#endif

__device__ __forceinline__ int iclamp(int v, int lo, int hi) {
    return v < lo ? lo : (v > hi ? hi : v);
}

// ---------------------------------------------------------------------------
// 1) Standard 3x3 conv, 256 -> 18 channels, stride 1, pad 1 (offset maps)
// ---------------------------------------------------------------------------
__global__ __launch_bounds__(256)
void offset_conv_kernel(const float* __restrict__ x,     // [NB, CIN, H, W]
                        const float* __restrict__ w,     // [18, CIN, 3, 3]
                        const float* __restrict__ bias,  // [18]
                        float* __restrict__ out,         // [NB, 18, H, W]
                        int H, int W)
{
    int idx = blockIdx.x * blockDim.x + threadIdx.x;
    int total = NB * 18 * H * W;
    if (idx >= total) return;
    int wq = idx % W; int t = idx / W;
    int hq = t % H;   t /= H;
    int o  = t % 18;  int b = t / 18;

    float acc = bias[o];
    const float* xb = x + (size_t)b * CIN * H * W;
    const float* wo = w + (size_t)o * CIN * 9;
    for (int c = 0; c < CIN; ++c) {
        const float* xc = xb + c * H * W;
        const float* wc = wo + c * 9;
        #pragma unroll
        for (int dy = -1; dy <= 1; ++dy) {
            int yy = hq + dy;
            if (yy < 0 || yy >= H) continue;
            #pragma unroll
            for (int dx = -1; dx <= 1; ++dx) {
                int xx = wq + dx;
                if (xx < 0 || xx >= W) continue;
                acc += xc[yy * W + xx] * wc[(dy + 1) * 3 + (dx + 1)];
            }
        }
    }
    out[idx] = acc;
}

// ---------------------------------------------------------------------------
// 2) Bilinear sampling metadata: per (b, tap, pos): 4 corner indices + 4 wgts
//    (shared across all 256 channels)
// ---------------------------------------------------------------------------
__global__ __launch_bounds__(256)
void deform_meta_kernel(const float* __restrict__ off,   // [NB, 18, H, W]
                        int*   __restrict__ midx,        // [NB, 9, H*W, 4]
                        float* __restrict__ mwgt,        // [NB, 9, H*W, 4]
                        int H, int W)
{
    int idx = blockIdx.x * blockDim.x + threadIdx.x;
    int npos = H * W;
    int total = NB * KTAPS * npos;
    if (idx >= total) return;
    int p = idx % npos; int t = idx / npos;
    int k = t % KTAPS;  int b = t / KTAPS;
    int hq = p / W, wq = p % W;

    float dy = off[((size_t)(b * 18 + 2 * k + 0)) * npos + p];
    float dx = off[((size_t)(b * 18 + 2 * k + 1)) * npos + p];
    float sy = (float)hq + (float)(k / 3 - 1) + dy;
    float sx = (float)wq + (float)(k % 3 - 1) + dx;
    float y0f = floorf(sy), x0f = floorf(sx);
    float ly = sy - y0f, lx = sx - x0f;
    int y0 = (int)y0f, x0 = (int)x0f;

    float wt[4] = { (1.f - ly) * (1.f - lx), (1.f - ly) * lx,
                    ly * (1.f - lx),         ly * lx };
    int ys[4] = { y0, y0, y0 + 1, y0 + 1 };
    int xs[4] = { x0, x0 + 1, x0, x0 + 1 };
    #pragma unroll
    for (int j = 0; j < 4; ++j) {
        bool valid = (ys[j] >= 0) && (ys[j] < H) && (xs[j] >= 0) && (xs[j] < W);
        midx[(size_t)idx * 4 + j] = iclamp(ys[j], 0, H - 1) * W + iclamp(xs[j], 0, W - 1);
        mwgt[(size_t)idx * 4 + j] = valid ? wt[j] : 0.0f;
    }
}

// ---------------------------------------------------------------------------
// 3) Deformable im2col for one batch: colbuf[c*9+tap, pos]
// ---------------------------------------------------------------------------
__global__ __launch_bounds__(256)
void deform_im2col_kernel(const float* __restrict__ xb,     // [CIN, npos] (batch slice)
                          const int*   __restrict__ midx,   // [9, npos, 4] (batch slice)
                          const float* __restrict__ mwgt,   // [9, npos, 4]
                          float* __restrict__ colbuf,       // [KTOT, npos]
                          int npos)
{
    int idx = blockIdx.x * blockDim.x + threadIdx.x;
    if (idx >= KTOT * npos) return;
    int p = idx % npos;
    int r = idx / npos;
    int c = r / KTAPS, tap = r % KTAPS;
    const int*   mi = midx + ((size_t)tap * npos + p) * 4;
    const float* mw = mwgt + ((size_t)tap * npos + p) * 4;
    const float* xc = xb + (size_t)c * npos;
    colbuf[idx] = mw[0] * xc[mi[0]] + mw[1] * xc[mi[1]]
                + mw[2] * xc[mi[2]] + mw[3] * xc[mi[3]];
}

// ---------------------------------------------------------------------------
// 4) WMMA f32 GEMM: Out[256, npos] = Wd[256, KTOT] x colbuf[KTOT, npos]
//    8 waves/WG; each wave owns a 16(M)x32(N) tile (one A fetch feeds 2 WMMAs);
//    B slab (64x32) staged into LDS via GLOBAL_LOAD_ASYNC_TO_LDS_B128.
// ---------------------------------------------------------------------------
__global__ __launch_bounds__(256)
void deform_gemm_wmma(const float* __restrict__ Wd,   // [COUT, KTOT]
                      const float* __restrict__ Bm,   // [KTOT, npos]
                      float* __restrict__ Out,        // [COUT, npos] (batch slice)
                      int npos)
{
    // Two 16-column halves; halves are 1024 floats apart (== 0 mod 64 banks),
    // so both keep the conflict-free 16-float-row bank pattern.
    __shared__ float lds[2][KS][16];              // 8 KB B slab (64 x 32)
    const int tid  = threadIdx.x;                 // 0..255
    const int lane = tid & 31;
    const int wave = tid >> 5;                    // 0..7
    const int m0   = blockIdx.x * 128 + wave * 16;
    const int n0   = blockIdx.y * 32;

    // A (16x4 f32): lanes 0-15 -> M rows, VGPR pair = K{0,1}; lanes 16-31 -> K{2,3}
    const int am = lane & 15;
    const int ak = (lane >> 4) * 2;
    // B (4x16 f32): mirrored layout, lane%16 = N column
    const int bn = lane & 15;
    const int bk = (lane >> 4) * 2;

    v8f acc0 = {}, acc1 = {};
    for (int k0 = 0; k0 < KTOT; k0 += KS) {
        // cooperative LDS fill: 64x32 = 2048 floats = 512 aligned 16B quads
        #pragma unroll
        for (int i = 0; i < 2; ++i) {
            int q  = tid + i * 256;               // 0..511
            int kk = q >> 3;                      // row 0..63
            int n4 = (q & 7) * 4;                 // quad col 0,4,...,28
            const float* g = &Bm[(size_t)(k0 + kk) * npos + (n0 + n4)];
            float* l = &lds[n4 >> 4][kk][n4 & 15];
#if HAVE_ASYNC_LDS
            __builtin_amdgcn_global_load_async_to_lds_b128(
                (AS1 gv4i*)g, (AS3 gv4i*)l, 0, 0);
#else
            *(float4*)l = *(const float4*)g;
#endif
        }
#if HAVE_ASYNC_LDS
        __builtin_amdgcn_s_wait_asynccnt(0);
#endif
        __syncthreads();

        const float* wrow = Wd + (size_t)(m0 + am) * KTOT + k0;
        __builtin_prefetch(wrow + KS, 0, 1);      // hint next weight slab

        #pragma unroll
        for (int ks = 0; ks < KS; ks += 4) {
            v2f a, b0, b1;
            a.x  = wrow[ks + ak];
            a.y  = wrow[ks + ak + 1];
            b0.x = lds[0][ks + bk][bn];
            b0.y = lds[0][ks + bk + 1][bn];
            b1.x = lds[1][ks + bk][bn];
            b1.y = lds[1][ks + bk + 1][bn];
            acc0 = __builtin_amdgcn_wmma_f32_16x16x4_f32(
                       false, a, false, b0, (short)0, acc0, false, false);
            acc1 = __builtin_amdgcn_wmma_f32_16x16x4_f32(
                       false, a, false, b1, (short)0, acc1, false, false);
        }
        __syncthreads();
    }

    // D layout: VGPR r -> M = r + 8*(lane>>4), N = lane&15
    const int mrow = m0 + ((lane >> 4) << 3);
    #pragma unroll
    for (int r = 0; r < 8; ++r) {
        Out[(size_t)(mrow + r) * npos + (n0 + bn)]      = acc0[r];
        Out[(size_t)(mrow + r) * npos + (n0 + 16 + bn)] = acc1[r];
    }
}

// ---------------------------------------------------------------------------
// Host launcher
// ---------------------------------------------------------------------------
extern "C" void kernel_launch(void* const* d_in, const int* in_sizes, int n_in,
                              void* d_out, int out_size, void* d_ws, size_t ws_size,
                              hipStream_t stream) {
    const float* kernel_x  = (const float*)d_in[0];   // [16,256,16,16]
    const float* search_x  = (const float*)d_in[1];   // [16,256,32,32]
    const float* Toffset_w = (const float*)d_in[2];   // [18,256,3,3]
    const float* Toffset_b = (const float*)d_in[3];   // [18]
    const float* Tdeform_w = (const float*)d_in[4];   // [256,256,3,3]
    const float* Soffset_w = (const float*)d_in[5];
    const float* Soffset_b = (const float*)d_in[6];
    const float* Sdeform_w = (const float*)d_in[7];

    float* out = (float*)d_out;
    float* out_k    = out;                    // [16,256,16,16]  1,048,576
    float* out_s    = out + 1048576;          // [16,256,32,32]  4,194,304
    float* out_koff = out + 5242880;          // [16,18,16,16]      73,728
    float* out_soff = out + 5316608;          // [16,18,32,32]     294,912

    // workspace partition (bytes, all 16B-aligned)
    char* ws = (char*)d_ws;
    int*   midxT = (int*)  (ws + 0);          // 16*9*256*4 ints   = 589,824 B
    float* mwgtT = (float*)(ws + 589824);     //                    589,824 B
    int*   midxS = (int*)  (ws + 1179648);    // 16*9*1024*4 ints = 2,359,296 B
    float* mwgtS = (float*)(ws + 3538944);    //                  2,359,296 B
    float* colbf = (float*)(ws + 5898240);    // 2304*1024 floats = 9,437,184 B

    const int NPK = 16 * 16;    // 256 positions, kernel branch
    const int NPS = 32 * 32;    // 1024 positions, search branch

    // 1) offset convolutions (outputs 3 & 4 of the tuple)
    offset_conv_kernel<<<(NB * 18 * NPK + 255) / 256, 256, 0, stream>>>(
        kernel_x, Toffset_w, Toffset_b, out_koff, 16, 16);
    offset_conv_kernel<<<(NB * 18 * NPS + 255) / 256, 256, 0, stream>>>(
        search_x, Soffset_w, Soffset_b, out_soff, 32, 32);

    // 2) bilinear metadata (channel-independent)
    deform_meta_kernel<<<(NB * KTAPS * NPK + 255) / 256, 256, 0, stream>>>(
        out_koff, midxT, mwgtT, 16, 16);
    deform_meta_kernel<<<(NB * KTAPS * NPS + 255) / 256, 256, 0, stream>>>(
        out_soff, midxS, mwgtS, 32, 32);

    // 3+4) per-batch deformable im2col + WMMA GEMM (stream order serializes colbf reuse)
    for (int b = 0; b < NB; ++b) {
        deform_im2col_kernel<<<(KTOT * NPK + 255) / 256, 256, 0, stream>>>(
            kernel_x + (size_t)b * CIN * NPK,
            midxT + (size_t)b * KTAPS * NPK * 4,
            mwgtT + (size_t)b * KTAPS * NPK * 4,
            colbf, NPK);
        deform_gemm_wmma<<<dim3(COUT / 128, NPK / 32), 256, 0, stream>>>(
            Tdeform_w, colbf, out_k + (size_t)b * COUT * NPK, NPK);
    }
    for (int b = 0; b < NB; ++b) {
        deform_im2col_kernel<<<(KTOT * NPS + 255) / 256, 256, 0, stream>>>(
            search_x + (size_t)b * CIN * NPS,
            midxS + (size_t)b * KTAPS * NPS * 4,
            mwgtS + (size_t)b * KTAPS * NPS * 4,
            colbf, NPS);
        deform_gemm_wmma<<<dim3(COUT / 128, NPS / 32), 256, 0, stream>>>(
            Sdeform_w, colbf, out_s + (size_t)b * COUT * NPS, NPS);
    }
}